// BroadcastConv_3152505995574
// MI455X (gfx1250) — compile-verified
//
#include <hip/hip_runtime.h>
#include <hip/hip_bf16.h>

#define AS1 __attribute__((address_space(1)))
#define AS3 __attribute__((address_space(3)))

// Async global->LDS copy (CDNA5 GLOBAL_LOAD_ASYNC_TO_LDS_B32, ASYNCcnt-tracked).
// Builtin signature (from hipcc diagnostic): (AS1 int*, AS3 int*, imm offset, imm cpol)
__device__ __forceinline__ void async_copy_b32(const float* gptr, float* lptr) {
#if __has_builtin(__builtin_amdgcn_global_load_async_to_lds_b32)
    __builtin_amdgcn_global_load_async_to_lds_b32((AS1 int*)gptr, (AS3 int*)lptr,
                                                  /*imm offset=*/0, /*cpol=*/0);
#else
    *lptr = *gptr;   // synchronous fallback (still correct)
#endif
}

__device__ __forceinline__ void wait_async_zero() {
#if __has_builtin(__builtin_amdgcn_s_wait_asynccnt)
    __builtin_amdgcn_s_wait_asynccnt(0);
#elif __has_builtin(__builtin_amdgcn_global_load_async_to_lds_b32)
    asm volatile("s_wait_asynccnt 0" ::: "memory");
#endif
}

// out[b,o,y,x] = max_{c,dy,dx} imgs[b,c,y+dy-1,x+dx-1] + kernel[o,c,2-dy,2-dx]
__global__ __launch_bounds__(256)
void maxplus_conv3x3_kernel(const float* __restrict__ imgs,
                            const float* __restrict__ kern,
                            float* __restrict__ out)
{
    constexpr int C     = 32;
    constexpr int HW    = 32;
    constexpr int TILE  = 34;             // 32 + 2 halo (-inf border)
    constexpr int TSZ   = TILE * TILE;    // 1156 floats per channel
    constexpr int CHUNK = 8;              // channels resident in LDS at once

    __shared__ float simg[CHUNK * TSZ];   // 9248 floats = 36,992 B
    __shared__ float sk[C * 9];           // 288 floats  =  1,152 B

    const int tid = threadIdx.x;
    const int o   = blockIdx.x;           // output channel
    const int b   = blockIdx.y;           // batch

    const float NEG_INF = -__builtin_inff();

    // One-time init: -inf everywhere (border stays -inf forever; interior is
    // overwritten by the async loads each chunk). Also stage this o's weights.
    for (int i = tid; i < CHUNK * TSZ; i += 256) simg[i] = NEG_INF;
    for (int t = tid; t < C * 9; t += 256)       sk[t]   = kern[o * C * 9 + t];

    // 4 output pixels per thread.
    float acc[4];
    int   py[4], px[4];
#pragma unroll
    for (int p = 0; p < 4; ++p) {
        acc[p] = NEG_INF;
        const int pix = tid + p * 256;
        py[p] = pix >> 5;
        px[p] = pix & 31;
    }

    __syncthreads();   // init visible to all waves before first async loads

    for (int c0 = 0; c0 < C; c0 += CHUNK) {
        // ---- stage CHUNK channels (4 KB/channel) into padded LDS tiles ----
        const float* gbase = imgs + (size_t)(b * C + c0) * (HW * HW);
#pragma unroll 4
        for (int it = 0; it < (CHUNK * HW * HW) / 256; ++it) {   // 32 iters
            const int e  = it * 256 + tid;     // flat element in this chunk
            const int ch = e >> 10;
            const int yy = (e >> 5) & 31;
            const int xx = e & 31;
            async_copy_b32(gbase + e,
                           &simg[ch * TSZ + (yy + 1) * TILE + (xx + 1)]);
        }
        wait_async_zero();     // my wave's async writes have landed in LDS
        __syncthreads();       // everyone's writes have landed

        // ---- accumulate max-plus over 8 channels x 9 taps ----
        for (int ch = 0; ch < CHUNK; ++ch) {
            const float* img = &simg[ch * TSZ];
            float w[9];
#pragma unroll
            for (int t = 0; t < 9; ++t)        // flipped kernel: tap t -> 8-t
                w[t] = sk[(c0 + ch) * 9 + (8 - t)];
#pragma unroll
            for (int p = 0; p < 4; ++p) {
                const float* base = img + py[p] * TILE + px[p];
#pragma unroll
                for (int dy = 0; dy < 3; ++dy)
#pragma unroll
                    for (int dx = 0; dx < 3; ++dx)
                        acc[p] = fmaxf(acc[p], base[dy * TILE + dx] + w[dy * 3 + dx]);
            }
        }
        __syncthreads();       // compute done before next chunk overwrites LDS
    }

    // ---- write out[b,o,:,:] ----
    float* obase = out + ((size_t)(b * 32 + o) << 10);
#pragma unroll
    for (int p = 0; p < 4; ++p)
        obase[tid + p * 256] = acc[p];
}

extern "C" void kernel_launch(void* const* d_in, const int* in_sizes, int n_in,
                              void* d_out, int out_size, void* d_ws, size_t ws_size,
                              hipStream_t stream) {
    const float* imgs = (const float*)d_in[0];   // (8,32,32,32) fp32
    const float* kern = (const float*)d_in[1];   // (32,32,3,3)  fp32
    float*       out  = (float*)d_out;           // (8,32,32,32) fp32

    dim3 grid(32, 8);   // (o, b)
    maxplus_conv3x3_kernel<<<grid, 256, 0, stream>>>(imgs, kern, out);
}